// PartGroupNet_5660766896172
// MI455X (gfx1250) — compile-verified
//
#include <hip/hip_runtime.h>
#include <math.h>

typedef __attribute__((ext_vector_type(2))) float v2f;
typedef __attribute__((ext_vector_type(8))) float v8f;

#define BS        64
#define CH        256
#define NPIX      3136      // 56*56
#define KPARTS    3
#define TILE      32        // pixels per LDS tile
#define ITERS     7         // tiles per block
#define BLKS_PER_B 14       // 14 * 7 * 32 = 3136
#define XS_STRIDE 36        // 32 + 4 pad: rows 16B aligned, banks spread (4c+j)

// ---------------------------------------------------------------------------
// Kernel 1: one pass over featmaps. Each block = (batch b, chunk of 224 pix).
// Stages x[ch][32] tiles in LDS, computes centers·x via V_WMMA_F32_16X16X4_F32
// (k padded 3->16), ||x||^2, 3-way softmax -> assign (written to d_out), and
// accumulates qx / sum_assign partials in registers; writes per-block partials
// to d_ws (deterministic, no atomics).
// ---------------------------------------------------------------------------
__global__ __launch_bounds__(256) void pgn_main(
    const float* __restrict__ x,        // [BS, CH, NPIX]
    const float* __restrict__ centers,  // [KPARTS, CH]
    const float* __restrict__ smooth,   // [KPARTS]
    float* __restrict__ assign_out,     // [BS, KPARTS, NPIX]
    float* __restrict__ qx_part,        // [BS, BLKS_PER_B, KPARTS, CH]
    float* __restrict__ sa_part)        // [BS, BLKS_PER_B, KPARTS]
{
    __shared__ float xs[CH * XS_STRIDE];        // 36,864 B
    __shared__ float centersT[CH * 16];         // 16,384 B (rows k>=3 zero)
    __shared__ float lds_cx[KPARTS * TILE];
    __shared__ float lds_xsq[TILE];
    __shared__ float lds_csq[KPARTS];
    __shared__ float lds_assign[KPARTS * TILE];

    const int tid  = threadIdx.x;
    const int blk  = blockIdx.x;                // 0..13
    const int b    = blockIdx.y;
    const int wv   = tid >> 5;
    const int lane = tid & 31;

    // centers, transposed + padded to 16 rows (A-matrix operand, conflict-free)
    for (int idx = tid; idx < CH * 16; idx += 256) {
        int c = idx >> 4, k = idx & 15;
        centersT[c * 16 + k] = (k < KPARTS) ? centers[k * CH + c] : 0.0f;
    }

    float a0 = 0.f, a1 = 0.f, a2 = 0.f;         // qx partials (c = tid)
    float sacc = 0.f;                            // sum_assign partial (tid<3)

    const float* xb = x + (size_t)b * CH * NPIX;

    for (int it = 0; it < ITERS; ++it) {
        const int n0 = (blk * ITERS + it) * TILE;

        // ---- Phase A: stage x tile [CH][TILE] into LDS (float4 streams) ----
        for (int idx = tid; idx < CH * (TILE / 4); idx += 256) {
            int c = idx >> 3, j4 = idx & 7;
            const float4 v =
                *reinterpret_cast<const float4*>(xb + (size_t)c * NPIX + n0 + j4 * 4);
            float* dst = &xs[c * XS_STRIDE + j4 * 4];
            dst[0] = v.x; dst[1] = v.y; dst[2] = v.z; dst[3] = v.w;
        }
        __syncthreads();

        // ---- Phase B: waves 0-1 do WMMA cx; wave 4 does ||x||^2; csq once --
#if __has_builtin(__builtin_amdgcn_wmma_f32_16x16x4_f32)
        if (wv < 2) {                             // uniform per wave: EXEC all 1s
            const int M    = lane & 15;           // A row / B col (N)
            const int koff = (lane >> 4) << 1;    // 0 or 2 (K half per ISA layout)
            const int j    = wv * 16 + M;         // output column in tile
            v8f acc = {0.f, 0.f, 0.f, 0.f, 0.f, 0.f, 0.f, 0.f};
            for (int c0 = 0; c0 < CH; c0 += 4) {
                v2f am, bm;
                am.x = centersT[(c0 + koff) * 16 + M];
                am.y = centersT[(c0 + koff + 1) * 16 + M];
                bm.x = xs[(c0 + koff) * XS_STRIDE + j];
                bm.y = xs[(c0 + koff + 1) * XS_STRIDE + j];
                acc = __builtin_amdgcn_wmma_f32_16x16x4_f32(
                    false, am, false, bm, (short)0, acc, false, false);
            }
            if (lane < 16) {                      // rows 0..2 live in lanes 0-15
                lds_cx[0 * TILE + j] = acc[0];
                lds_cx[1 * TILE + j] = acc[1];
                lds_cx[2 * TILE + j] = acc[2];
            }
        }
#else
        if (tid < TILE) {
            float d0 = 0.f, d1 = 0.f, d2 = 0.f;
            for (int c = 0; c < CH; ++c) {
                float xv = xs[c * XS_STRIDE + tid];
                d0 += centersT[c * 16 + 0] * xv;
                d1 += centersT[c * 16 + 1] * xv;
                d2 += centersT[c * 16 + 2] * xv;
            }
            lds_cx[0 * TILE + tid] = d0;
            lds_cx[1 * TILE + tid] = d1;
            lds_cx[2 * TILE + tid] = d2;
        }
#endif
        if (tid >= 128 && tid < 128 + TILE) {     // wave 4: ||x||^2 per column
            int j = tid - 128;
            float s = 0.f;
            for (int c = 0; c < CH; ++c) {
                float v = xs[c * XS_STRIDE + j];
                s += v * v;
            }
            lds_xsq[j] = s;
        }
        if (it == 0 && tid >= 192 && tid < 192 + KPARTS) {
            int k = tid - 192;
            float s = 0.f;
            for (int c = 0; c < CH; ++c) {
                float v = centersT[c * 16 + k];
                s += v * v;
            }
            lds_csq[k] = s;
        }
        __syncthreads();

        // ---- Phase C: 3-way softmax per column (wave 0) -> assign ----------
        if (tid < TILE) {
            const int j = tid;
            const float xsq = lds_xsq[j];
            float beta0 = 1.f / (1.f + __expf(-smooth[0]));
            float beta1 = 1.f / (1.f + __expf(-smooth[1]));
            float beta2 = 1.f / (1.f + __expf(-smooth[2]));
            float l0 = fminf(2.f * lds_cx[0 * TILE + j] - xsq - lds_csq[0], 0.f) / beta0;
            float l1 = fminf(2.f * lds_cx[1 * TILE + j] - xsq - lds_csq[1], 0.f) / beta1;
            float l2 = fminf(2.f * lds_cx[2 * TILE + j] - xsq - lds_csq[2], 0.f) / beta2;
            float mx = fmaxf(l0, fmaxf(l1, l2));
            float e0 = __expf(l0 - mx), e1 = __expf(l1 - mx), e2 = __expf(l2 - mx);
            float inv = 1.f / (e0 + e1 + e2);
            float as0 = e0 * inv, as1 = e1 * inv, as2 = e2 * inv;
            lds_assign[0 * TILE + j] = as0;
            lds_assign[1 * TILE + j] = as1;
            lds_assign[2 * TILE + j] = as2;
            size_t base = ((size_t)b * KPARTS) * NPIX + n0 + j;
            assign_out[base]            = as0;
            assign_out[base + NPIX]     = as1;
            assign_out[base + 2 * NPIX] = as2;
        }
        __syncthreads();

        // ---- Phase D: qx partial accumulation (c = tid, reuse LDS tile) ----
        {
            const int c = tid;
            for (int j = 0; j < TILE; ++j) {
                float xv = xs[c * XS_STRIDE + j];
                a0 += lds_assign[0 * TILE + j] * xv;
                a1 += lds_assign[1 * TILE + j] * xv;
                a2 += lds_assign[2 * TILE + j] * xv;
            }
        }
        if (tid < KPARTS) {
            float s = 0.f;
            for (int j = 0; j < TILE; ++j) s += lds_assign[tid * TILE + j];
            sacc += s;
        }
        __syncthreads();   // protect xs/lds_assign before next tile overwrite
    }

    // ---- Write per-block partials (deterministic reduction in kernel 2) ----
    float* qp = qx_part + ((size_t)(b * BLKS_PER_B + blk) * KPARTS) * CH;
    qp[0 * CH + tid] = a0;
    qp[1 * CH + tid] = a1;
    qp[2 * CH + tid] = a2;
    if (tid < KPARTS)
        sa_part[(b * BLKS_PER_B + blk) * KPARTS + tid] = sacc;
}

// ---------------------------------------------------------------------------
// Kernel 2: per-(b,k) finalize — reduce partials, center/scale, L2-normalize
// over channels, write outputs transposed to [BS, CH, KPARTS].
// ---------------------------------------------------------------------------
__global__ __launch_bounds__(256) void pgn_finalize(
    const float* __restrict__ centers,
    const float* __restrict__ smooth,
    const float* __restrict__ qx_part,
    const float* __restrict__ sa_part,
    float* __restrict__ outputs)        // [BS, CH, KPARTS]
{
    __shared__ float red[256];
    const int bk = blockIdx.x;
    const int b = bk / KPARTS, k = bk % KPARTS;
    const int c = threadIdx.x;

    float q = 0.f, sa = 0.f;
    for (int t = 0; t < BLKS_PER_B; ++t) {
        q  += qx_part[((size_t)(b * BLKS_PER_B + t) * KPARTS + k) * CH + c];
        sa += sa_part[(b * BLKS_PER_B + t) * KPARTS + k];
    }
    sa = fmaxf(sa, 1e-5f);

    float beta  = 1.f / (1.f + __expf(-smooth[k]));
    float sigma = sqrtf(beta * 0.5f);
    float o = (q / sa - centers[k * CH + c]) / sigma;

    red[c] = o * o;
    __syncthreads();
    for (int s = 128; s > 0; s >>= 1) {
        if (c < s) red[c] += red[c + s];
        __syncthreads();
    }
    float nrm = fmaxf(sqrtf(red[0]), 1e-12f);
    outputs[((size_t)b * CH + c) * KPARTS + k] = o / nrm;
}

// ---------------------------------------------------------------------------
extern "C" void kernel_launch(void* const* d_in, const int* in_sizes, int n_in,
                              void* d_out, int out_size, void* d_ws, size_t ws_size,
                              hipStream_t stream) {
    const float* x       = (const float*)d_in[0];   // featmaps [64,256,56,56]
    const float* centers = (const float*)d_in[1];   // [3,256]
    const float* smooth  = (const float*)d_in[2];   // [3]

    float* out     = (float*)d_out;
    float* outputs = out;                           // [64,256,3]
    float* assign  = out + (size_t)BS * CH * KPARTS;// [64,3,56,56]

    float* qx_part = (float*)d_ws;                              // [64,14,3,256]
    float* sa_part = qx_part + (size_t)BS * BLKS_PER_B * KPARTS * CH;

    dim3 g1(BLKS_PER_B, BS);
    pgn_main<<<g1, 256, 0, stream>>>(x, centers, smooth, assign, qx_part, sa_part);
    pgn_finalize<<<BS * KPARTS, 256, 0, stream>>>(centers, smooth, qx_part, sa_part,
                                                  outputs);
}